// GraphSAGE_Structural_74577812128601
// MI455X (gfx1250) — compile-verified
//
#include <hip/hip_runtime.h>
#include <hip/hip_bf16.h>

#define NNODES 100000
#define NEDGES 1600000
#define FIN    128
#define HDIM   64
#define CFDIM  16
#define NGRAPH 256

typedef __attribute__((ext_vector_type(2))) float v2f;
typedef __attribute__((ext_vector_type(8))) float v8f;

#define ORD_NEG_INF 0x007FFFFFu  // f2ord(-inf)

// monotone map: float total order -> unsigned order (for atomic max on floats)
__device__ __forceinline__ unsigned f2ord(float f) {
  unsigned u = __float_as_uint(f);
  return (u & 0x80000000u) ? ~u : (u | 0x80000000u);
}
__device__ __forceinline__ float ord2f(unsigned u) {
  unsigned b = (u & 0x80000000u) ? (u ^ 0x80000000u) : ~u;
  return __uint_as_float(b);
}
__device__ __forceinline__ float fixup(float v) {
  // mirror jnp.where(isfinite(v), v, 0): NaN/±inf -> 0
  return (fabsf(v) < __builtin_inff()) ? v : 0.0f;
}

// ---------------------------------------------------------------- fill
__global__ void k_fill_u32(unsigned* __restrict__ p, unsigned v, int n) {
  int i = blockIdx.x * blockDim.x + threadIdx.x;
  if (i < n) p[i] = v;
}

// ------------------------------------------------- edge scatter-max (segment_max)
// one thread handles 4 features of one edge; agg is ord-encoded u32.
// F==128: one wave == one edge -> scalarize the edge-index fetch (s_load path)
// and prefetch the streaming index arrays ahead (global_prefetch_b8).
template <int F>
__global__ void k_scatter_max(const float* __restrict__ feat,
                              const long long* __restrict__ src,
                              const long long* __restrict__ dst,
                              unsigned* __restrict__ agg) {
  constexpr int FQ = F / 4;
  int t = blockIdx.x * blockDim.x + threadIdx.x;
  if (t >= NEDGES * FQ) return;
  int e, c;
  if constexpr (FQ == 32) {
    // wave-uniform edge id: provably uniform -> scalar loads of src/dst
    e = __builtin_amdgcn_readfirstlane(t >> 5);
    c = threadIdx.x & 31;
    if (c == 0) {
      __builtin_prefetch(src + e + 4096, 0, 0);
      __builtin_prefetch(dst + e + 4096, 0, 0);
    }
  } else {
    e = t / FQ;
    c = t - e * FQ;
    if ((threadIdx.x & 31) == 0) {
      __builtin_prefetch(src + (t >> 5) * (32 / FQ) + 4096, 0, 0);
      __builtin_prefetch(dst + (t >> 5) * (32 / FQ) + 4096, 0, 0);
    }
  }
  int s = (int)src[e];
  int d = (int)dst[e];
  float4 v = *(const float4*)(feat + (size_t)s * F + c * 4);
  unsigned* ap = agg + (size_t)d * F + c * 4;
  atomicMax(ap + 0, f2ord(v.x));
  atomicMax(ap + 1, f2ord(v.y));
  atomicMax(ap + 2, f2ord(v.z));
  atomicMax(ap + 3, f2ord(v.w));
}

// ------------------------------------------------- fused SAGE layer (WMMA fp32)
// out[n,0:64] = relu( fixup(decode(agg[n,:])) @ Wl + bl + xin[n,:] @ Wr )
// block = 256 threads = 8 waves; each wave owns a 16x16 output tile of a
// 32(node) x 64(col) block tile. K processed in 64-wide LDS-staged chunks.
// Two independent accumulator chains (agg@Wl vs x@Wr) halve the XDL RAW chain.
#define KC 64
__global__ void __launch_bounds__(256)
k_sage_wmma(const unsigned* __restrict__ agg,   // [N,K] ord-encoded
            const float* __restrict__ xin,      // [N,K]
            const float* __restrict__ Wl,       // [K,64] row-major
            const float* __restrict__ bl,       // [64]
            const float* __restrict__ Wr,       // [K,64]
            float* __restrict__ out,            // [N,64]
            int K) {
  __shared__ float sWl[KC * 66];   // +2 pad: kills half-wave bank conflicts
  __shared__ float sWr[KC * 66];
  __shared__ float sA[32 * 65];    // +1 pad: rotates banks per row
  __shared__ float sX[32 * 65];

  const int tid  = threadIdx.x;
  const int lane = tid & 31;
  const int wave = tid >> 5;
  const int nb   = blockIdx.x * 32;

  const int col = lane & 15;       // N-col within tile / A-row within half
  const int hi  = lane >> 4;       // lane half selects K-pair (ISA A/B layout)
  const int n0  = (wave & 3) * 16; // which 16 output cols
  const int mh  = (wave >> 2) * 16;// which 16 of the 32 rows
  const int row = mh + col;        // A row in [0,32)

  v8f accL, accR;
#pragma unroll
  for (int i = 0; i < 8; ++i) { accL[i] = 0.0f; accR[i] = 0.0f; }

  for (int kc = 0; kc < K; kc += KC) {
    // stage weight chunk [KC x 64] for both matrices (coalesced)
    for (int idx = tid; idx < KC * 64; idx += 256) {
      int k = idx >> 6, n = idx & 63;
      sWl[k * 66 + n] = Wl[(size_t)(kc + k) * 64 + n];
      sWr[k * 66 + n] = Wr[(size_t)(kc + k) * 64 + n];
    }
    // stage node tile [32 x KC]: decoded agg (with isfinite->0) and raw x
    for (int idx = tid; idx < 32 * KC; idx += 256) {
      int r = idx >> 6, k = idx & 63;
      size_t g = (size_t)(nb + r) * K + kc + k;
      sA[r * 65 + k] = fixup(ord2f(agg[g]));
      sX[r * 65 + k] = xin[g];
    }
    __syncthreads();

#pragma unroll
    for (int k0 = 0; k0 < KC; k0 += 4) {
      int ka = k0 + hi * 2;  // lanes 0-15 -> K={k0,k0+1}; lanes 16-31 -> {k0+2,k0+3}
      v2f aA, aX, bL, bR;
      aA[0] = sA[row * 65 + ka];       aA[1] = sA[row * 65 + ka + 1];
      aX[0] = sX[row * 65 + ka];       aX[1] = sX[row * 65 + ka + 1];
      bL[0] = sWl[ka * 66 + n0 + col]; bL[1] = sWl[(ka + 1) * 66 + n0 + col];
      bR[0] = sWr[ka * 66 + n0 + col]; bR[1] = sWr[(ka + 1) * 66 + n0 + col];
      accL = __builtin_amdgcn_wmma_f32_16x16x4_f32(false, aA, false, bL,
                                                   (short)0, accL, false, false);
      accR = __builtin_amdgcn_wmma_f32_16x16x4_f32(false, aX, false, bR,
                                                   (short)0, accR, false, false);
    }
    __syncthreads();
  }

  // C/D layout: vgpr i, lane half hi -> row (i + 8*hi), col = lane&15
  const float bias = bl[n0 + col];
#pragma unroll
  for (int i = 0; i < 8; ++i) {
    int m = mh + i + 8 * hi;
    float v = accL[i] + accR[i] + bias;
    out[(size_t)(nb + m) * 64 + n0 + col] = v > 0.0f ? v : 0.0f;
  }
}

// ------------------------------------------------- attention pooling pieces
__global__ void k_closeness(const float* __restrict__ clo,
                            const float* __restrict__ Wc,
                            const float* __restrict__ bc,
                            const long long* __restrict__ batch,
                            float* __restrict__ c,
                            unsigned* __restrict__ cmax_ord) {
  int n = blockIdx.x * blockDim.x + threadIdx.x;
  if (n >= NNODES) return;
  float s = bc[0];
#pragma unroll
  for (int f = 0; f < CFDIM; ++f) s += clo[(size_t)n * CFDIM + f] * Wc[f];
  c[n] = s;
  atomicMax(cmax_ord + (int)batch[n], f2ord(s));
}

__global__ void k_expsum(const float* __restrict__ c,
                         const long long* __restrict__ batch,
                         const unsigned* __restrict__ cmax_ord,
                         float* __restrict__ e,
                         float* __restrict__ esum,
                         float* __restrict__ cnt) {
  int n = blockIdx.x * blockDim.x + threadIdx.x;
  if (n >= NNODES) return;
  int g = (int)batch[n];
  float ev = expf(c[n] - ord2f(cmax_ord[g]));
  e[n] = ev;
  atomicAdd(esum + g, ev);
  atomicAdd(cnt + g, 1.0f);
}

__global__ void k_pool(const float* __restrict__ h,
                       const long long* __restrict__ batch,
                       const float* __restrict__ e,
                       const float* __restrict__ esum,
                       const float* __restrict__ cnt,
                       unsigned* __restrict__ pooled_ord) {
  int t = blockIdx.x * blockDim.x + threadIdx.x;
  if (t >= NNODES * 16) return;
  int n = t >> 4, cq = t & 15;
  int g = (int)batch[n];
  float p = e[n] / esum[g] * cnt[g];
  float4 v = *(const float4*)(h + (size_t)n * 64 + cq * 4);
  unsigned* pp = pooled_ord + g * 64 + cq * 4;
  atomicMax(pp + 0, f2ord(p * v.x));
  atomicMax(pp + 1, f2ord(p * v.y));
  atomicMax(pp + 2, f2ord(p * v.z));
  atomicMax(pp + 3, f2ord(p * v.w));
}

__global__ void k_head(const unsigned* __restrict__ pooled_ord,
                       const float* __restrict__ Wa1,
                       const float* __restrict__ ba1,
                       const float* __restrict__ Wa2,
                       const float* __restrict__ ba2,
                       float* __restrict__ outp) {
  int g = threadIdx.x;  // one block of 256
  if (g >= NGRAPH) return;
  float pool[64];
#pragma unroll
  for (int f = 0; f < 64; ++f) pool[f] = fixup(ord2f(pooled_ord[g * 64 + f]));
  float o = ba2[0];
  for (int j = 0; j < 16; ++j) {
    float s = ba1[j];
#pragma unroll
    for (int f = 0; f < 64; ++f) s += pool[f] * Wa1[f * 16 + j];
    o += (s > 0.0f ? s : 0.0f) * Wa2[j];
  }
  outp[g] = o;
}

// ---------------------------------------------------------------- launch
extern "C" void kernel_launch(void* const* d_in, const int* in_sizes, int n_in,
                              void* d_out, int out_size, void* d_ws, size_t ws_size,
                              hipStream_t stream) {
  const float*     x     = (const float*)d_in[0];
  const long long* ei    = (const long long*)d_in[1];
  const long long* batch = (const long long*)d_in[2];
  const float*     clo   = (const float*)d_in[3];
  const float *W1l = (const float*)d_in[4],  *b1l = (const float*)d_in[5],  *W1r = (const float*)d_in[6];
  const float *W2l = (const float*)d_in[7],  *b2l = (const float*)d_in[8],  *W2r = (const float*)d_in[9];
  const float *W3l = (const float*)d_in[10], *b3l = (const float*)d_in[11], *W3r = (const float*)d_in[12];
  const float *Wc  = (const float*)d_in[13], *bc  = (const float*)d_in[14];
  const float *Wa1 = (const float*)d_in[15], *ba1 = (const float*)d_in[16];
  const float *Wa2 = (const float*)d_in[17], *ba2 = (const float*)d_in[18];
  float* out = (float*)d_out;

  char* ws = (char*)d_ws;
  size_t off = 0;
  auto alloc = [&](size_t bytes) -> void* {
    void* p = ws + off;
    off += (bytes + 255) & ~(size_t)255;
    return p;
  };
  unsigned* agg    = (unsigned*)alloc((size_t)NNODES * FIN * 4);
  float*    hA     = (float*)   alloc((size_t)NNODES * HDIM * 4);
  float*    hB     = (float*)   alloc((size_t)NNODES * HDIM * 4);
  float*    cbuf   = (float*)   alloc((size_t)NNODES * 4);
  float*    ebuf   = (float*)   alloc((size_t)NNODES * 4);
  unsigned* cmax   = (unsigned*)alloc((size_t)NGRAPH * 4);
  float*    esum   = (float*)   alloc((size_t)NGRAPH * 4);
  float*    cnt    = (float*)   alloc((size_t)NGRAPH * 4);
  unsigned* pooled = (unsigned*)alloc((size_t)NGRAPH * HDIM * 4);

  const long long* src = ei;
  const long long* dst = ei + NEDGES;

  auto blocks = [](long long n) { return dim3((unsigned)((n + 255) / 256)); };

  // ---- layer 1 (K=128)
  k_fill_u32<<<blocks((long long)NNODES * FIN), 256, 0, stream>>>(agg, ORD_NEG_INF, NNODES * FIN);
  k_scatter_max<FIN><<<blocks((long long)NEDGES * (FIN / 4)), 256, 0, stream>>>(x, src, dst, agg);
  k_sage_wmma<<<NNODES / 32, 256, 0, stream>>>(agg, x, W1l, b1l, W1r, hA, FIN);

  // ---- layer 2 (K=64)
  k_fill_u32<<<blocks((long long)NNODES * HDIM), 256, 0, stream>>>(agg, ORD_NEG_INF, NNODES * HDIM);
  k_scatter_max<HDIM><<<blocks((long long)NEDGES * (HDIM / 4)), 256, 0, stream>>>(hA, src, dst, agg);
  k_sage_wmma<<<NNODES / 32, 256, 0, stream>>>(agg, hA, W2l, b2l, W2r, hB, HDIM);

  // ---- layer 3 (K=64)
  k_fill_u32<<<blocks((long long)NNODES * HDIM), 256, 0, stream>>>(agg, ORD_NEG_INF, NNODES * HDIM);
  k_scatter_max<HDIM><<<blocks((long long)NEDGES * (HDIM / 4)), 256, 0, stream>>>(hB, src, dst, agg);
  k_sage_wmma<<<NNODES / 32, 256, 0, stream>>>(agg, hB, W3l, b3l, W3r, hA, HDIM);

  // ---- attention pooling
  k_fill_u32<<<1, 256, 0, stream>>>(cmax, ORD_NEG_INF, NGRAPH);
  k_fill_u32<<<1, 256, 0, stream>>>((unsigned*)esum, 0u, NGRAPH);
  k_fill_u32<<<1, 256, 0, stream>>>((unsigned*)cnt, 0u, NGRAPH);
  k_fill_u32<<<blocks((long long)NGRAPH * HDIM), 256, 0, stream>>>(pooled, ORD_NEG_INF, NGRAPH * HDIM);

  k_closeness<<<blocks(NNODES), 256, 0, stream>>>(clo, Wc, bc, batch, cbuf, cmax);
  k_expsum<<<blocks(NNODES), 256, 0, stream>>>(cbuf, batch, cmax, ebuf, esum, cnt);
  k_pool<<<blocks((long long)NNODES * 16), 256, 0, stream>>>(hA, batch, ebuf, esum, cnt, pooled);
  k_head<<<1, 256, 0, stream>>>(pooled, Wa1, ba1, Wa2, ba2, out);
}